// InterpolationNetwork_12833362280479
// MI455X (gfx1250) — compile-verified
//
#include <hip/hip_runtime.h>

#define B_ 2
#define T_ 24
#define F_ 8
#define H_ 48
#define W_ 48
#define R_ 64
#define N_ (B_*H_*W_)          // 4608
#define WPB 4                   // waves (sites) per block
#define Y_ELEMS ((long)B_*R_*3*F_*H_*W_)   // 7,077,888

typedef float v2f __attribute__((ext_vector_type(2)));
typedef float v8f __attribute__((ext_vector_type(8)));

__device__ __forceinline__ float sigmoid_(float v) { return 1.0f / (1.0f + expf(-v)); }

__global__ void zero_accum_k(float* a) { a[0] = 0.0f; a[1] = 0.0f; }
__global__ void finalize_k(const float* __restrict__ a, float* __restrict__ mse) { *mse = a[0] / a[1]; }

// ---------------- Interpolation path: one wave per spatial site n ----------------
__global__ __launch_bounds__(128) void interp_k(
    const float* __restrict__ x, const float* __restrict__ obs,
    const float* __restrict__ ker, const float* __restrict__ cci,
    float* __restrict__ out)
{
  __shared__ float sAlpha[T_];
  __shared__ float sCci[F_][16];       // zero-padded B operand: no predication in WMMA loop
  __shared__ float sX[WPB][F_][T_];
  __shared__ float sG[WPB][F_][T_];
  __shared__ float sYs[WPB][F_][R_];   // sigmoid(y)
  __shared__ float sWs[WPB][F_][R_];   // sigmoid(den)
  __shared__ float sYt[WPB][F_][R_];   // sigmoid(y_trans)
  __shared__ float sS [WPB][R_][F_];   // sm * (ys - mean)  (GEMM A operand)
  __shared__ float sMean[WPB][F_];

  const int tid = threadIdx.x;
  const int wv  = tid >> 5;
  const int ln  = tid & 31;
  const int n   = blockIdx.x * WPB + wv;
  const int b   = n / (H_ * W_);
  const int hw  = n - b * (H_ * W_);
  const int h   = hw / W_;
  const int w   = hw - h * W_;

  if (tid < T_) sAlpha[tid] = log1pf(expf(ker[tid]));   // softplus
  {
    const int k = tid >> 4, c = tid & 15;               // 128 threads cover 8x16
    sCci[k][c] = (c < F_) ? cci[k * F_ + c] : 0.0f;
  }

  for (int i = ln; i < F_ * T_; i += 32) {
    const int f = i / T_;
    const int t = i - f * T_;
    const int xi = (((b * T_ + t) * F_ + f) * H_ + h) * W_ + w;
    sX[wv][f][t] = x[xi];
    sG[wv][f][t] = (obs[xi] > 0.0f) ? 1.0f : 0.0f;
  }
  __syncthreads();

  // Phase 1: per (f,o) stable logsumexp kernel regression (16 pairs / lane)
  for (int p = ln; p < F_ * R_; p += 32) {
    const int f = p >> 6;
    const int o = p & 63;
    const float ro = (float)o * (64.0f / 63.0f);   // linspace(0, 64, 64)
    float mx = -INFINITY;
    #pragma unroll
    for (int t = 0; t < T_; ++t) {
      if (sG[wv][f][t] > 0.0f) {
        const float dd = (float)t - ro;
        mx = fmaxf(mx, -sAlpha[t] * dd * dd);
      }
    }
    float sd1 = 0.f, sn1 = 0.f, sd10 = 0.f, sn10 = 0.f;
    #pragma unroll
    for (int t = 0; t < T_; ++t) {
      if (sG[wv][f][t] > 0.0f) {
        const float dd = (float)t - ro;
        const float l1 = -sAlpha[t] * dd * dd - mx;   // <= 0
        const float e1  = expf(l1);
        const float e10 = expf(10.0f * l1);           // max10 == 10*max1
        const float xv = sX[wv][f][t];
        sd1  += e1;  sn1  += e1  * xv;
        sd10 += e10; sn10 += e10 * xv;
      }
    }
    sYs[wv][f][o] = sigmoid_(sn1 / sd1);
    sWs[wv][f][o] = sigmoid_(mx + logf(sd1));
    sYt[wv][f][o] = sigmoid_(sn10 / sd10);
  }
  __syncthreads();

  // Phase 2a: per-f mean over the 64 outputs
  if (ln < F_) {
    float s = 0.f;
    for (int o = 0; o < R_; ++o) s += sYs[wv][ln][o];
    sMean[wv][ln] = s * (1.0f / R_);
  }
  __syncthreads();

  // Phase 2b: per-o softmax over F, build GEMM operand S[o][f]
  for (int o = ln; o < R_; o += 32) {
    float wvv[F_];
    float wm = -INFINITY;
    #pragma unroll
    for (int f = 0; f < F_; ++f) { wvv[f] = sWs[wv][f][o]; wm = fmaxf(wm, wvv[f]); }
    float se = 0.f;
    #pragma unroll
    for (int f = 0; f < F_; ++f) { wvv[f] = expf(wvv[f] - wm); se += wvv[f]; }
    const float inv = 1.0f / se;
    #pragma unroll
    for (int f = 0; f < F_; ++f)
      sS[wv][o][f] = (wvv[f] * inv) * (sYs[wv][f][o] - sMean[wv][f]);
  }
  __syncthreads();

  // Phase 3: W2(64x8) = S(64x8) @ cci(8x8) via fp32 WMMA 16x16x4, then outputs.
  const int col   = ln & 15;           // A: M row within tile / B,D: N column
  const int khalf = (ln >> 4) * 2;     // lanes 16-31 carry K=2,3 halves
  const long strideC = (long)F_ * H_ * W_;
  #pragma unroll
  for (int tile = 0; tile < 4; ++tile) {
    v8f c = {0.f, 0.f, 0.f, 0.f, 0.f, 0.f, 0.f, 0.f};
    #pragma unroll
    for (int kk = 0; kk < 2; ++kk) {
      const int kb = kk * 4 + khalf;           // k base for this lane half (0..7 = f)
      const int arow = tile * 16 + col;
      v2f a, bf;
      a.x  = sS[wv][arow][kb];
      a.y  = sS[wv][arow][kb + 1];
      bf.x = sCci[kb][col];                    // padded: unconditional loads
      bf.y = sCci[kb + 1][col];
      c = __builtin_amdgcn_wmma_f32_16x16x4_f32(false, a, false, bf,
                                                (short)0, c, false, false);
    }
    // D layout: lane<16 -> N=lane, M=i ; lane>=16 -> N=lane-16, M=8+i
    if (col < F_) {
      const int fp = col;
      const float mn = sMean[wv][fp];
      #pragma unroll
      for (int i = 0; i < 8; ++i) {
        const int o = tile * 16 + ((ln < 16) ? i : (8 + i));
        const float rep = c[i] + mn;                       // rep1[fp][o]
        const float o0 = sigmoid_(rep);
        const float o1 = sigmoid_(expf(sWs[wv][fp][o]));   // intensity
        const float o2 = sigmoid_(sYt[wv][fp][o] - rep);   // y_trans - rep1
        long base = ((((long)b * R_ + o) * 3 + 0) * F_ + fp);
        base = (base * H_ + h) * W_ + w;
        out[base]               = o0;
        out[base + strideC]     = o1;
        out[base + 2 * strideC] = o2;
      }
    }
  }
}

// ---------------- Reconstruction path + MSE: one wave per site n ----------------
__global__ __launch_bounds__(128) void recon_k(
    const float* __restrict__ x, const float* __restrict__ obs,
    const float* __restrict__ rnd, const float* __restrict__ ker,
    const float* __restrict__ cci, float* __restrict__ accum)
{
  __shared__ float sAlpha[T_];
  __shared__ float sCci[F_][16];        // zero-padded B operand
  __shared__ float sX[WPB][F_][T_];
  __shared__ float sGh[WPB][F_][T_];    // hold gate (rand>0.2 && m>0)
  __shared__ float sHeld[WPB][F_][T_];  // held gate (m>0 && rand<=0.2)
  __shared__ float sYs[WPB][F_][T_];
  __shared__ float sWs[WPB][F_][T_];
  __shared__ float sS [WPB][32][F_];    // rows 24..31 zero-padded for WMMA M-tiles
  __shared__ float sMean[WPB][F_];
  __shared__ float sRed[128];

  const int tid = threadIdx.x;
  const int wv  = tid >> 5;
  const int ln  = tid & 31;
  const int n   = blockIdx.x * WPB + wv;
  const int b   = n / (H_ * W_);
  const int hw  = n - b * (H_ * W_);
  const int h   = hw / W_;
  const int w   = hw - h * W_;

  if (tid < T_) sAlpha[tid] = log1pf(expf(ker[tid]));
  {
    const int k = tid >> 4, c = tid & 15;
    sCci[k][c] = (c < F_) ? cci[k * F_ + c] : 0.0f;
  }

  for (int i = ln; i < F_ * T_; i += 32) {
    const int f = i / T_;
    const int t = i - f * T_;
    const int xi = (((b * T_ + t) * F_ + f) * H_ + h) * W_ + w;
    const float m  = (obs[xi] > 0.0f) ? 1.0f : 0.0f;
    const float rv = rnd[(n * F_ + f) * T_ + t];
    sX[wv][f][t]    = x[xi];
    sGh[wv][f][t]   = (m > 0.0f && rv > 0.2f)  ? 1.0f : 0.0f;
    sHeld[wv][f][t] = (m > 0.0f && rv <= 0.2f) ? 1.0f : 0.0f;
  }
  __syncthreads();

  // Phase 1: per (f,o), o in [0,T): ref_o = o (reconstruction mode)
  for (int p = ln; p < F_ * T_; p += 32) {
    const int f = p / T_;
    const int o = p - f * T_;
    float mx = -INFINITY;
    #pragma unroll
    for (int t = 0; t < T_; ++t) {
      if (sGh[wv][f][t] > 0.0f) {
        const float dd = (float)(t - o);
        mx = fmaxf(mx, -sAlpha[t] * dd * dd);
      }
    }
    float sd = 0.f, sn = 0.f;
    #pragma unroll
    for (int t = 0; t < T_; ++t) {
      if (sGh[wv][f][t] > 0.0f) {
        const float dd = (float)(t - o);
        const float e = expf(-sAlpha[t] * dd * dd - mx);
        sd += e; sn += e * sX[wv][f][t];
      }
    }
    sYs[wv][f][o] = sigmoid_(sn / sd);
    sWs[wv][f][o] = sigmoid_(mx + logf(sd));
  }
  __syncthreads();

  if (ln < F_) {
    float s = 0.f;
    for (int o = 0; o < T_; ++o) s += sYs[wv][ln][o];
    sMean[wv][ln] = s * (1.0f / T_);
  }
  __syncthreads();

  // Phase 2b: one o per lane (o = ln); rows >= T_ zero-padded
  {
    const int o = ln;
    if (o < T_) {
      float wvv[F_];
      float wm = -INFINITY;
      #pragma unroll
      for (int f = 0; f < F_; ++f) { wvv[f] = sWs[wv][f][o]; wm = fmaxf(wm, wvv[f]); }
      float se = 0.f;
      #pragma unroll
      for (int f = 0; f < F_; ++f) { wvv[f] = expf(wvv[f] - wm); se += wvv[f]; }
      const float inv = 1.0f / se;
      #pragma unroll
      for (int f = 0; f < F_; ++f)
        sS[wv][o][f] = (wvv[f] * inv) * (sYs[wv][f][o] - sMean[wv][f]);
    } else {
      #pragma unroll
      for (int f = 0; f < F_; ++f) sS[wv][o][f] = 0.0f;
    }
  }
  __syncthreads();

  // Phase 3: W2(24x8, padded to 32) = S @ cci_r via fp32 WMMA; accumulate held MSE
  const int col   = ln & 15;
  const int khalf = (ln >> 4) * 2;
  float ls = 0.f, lc = 0.f;
  #pragma unroll
  for (int tile = 0; tile < 2; ++tile) {
    v8f c = {0.f, 0.f, 0.f, 0.f, 0.f, 0.f, 0.f, 0.f};
    #pragma unroll
    for (int kk = 0; kk < 2; ++kk) {
      const int kb = kk * 4 + khalf;
      v2f a, bf;
      a.x  = sS[wv][tile * 16 + col][kb];
      a.y  = sS[wv][tile * 16 + col][kb + 1];
      bf.x = sCci[kb][col];
      bf.y = sCci[kb + 1][col];
      c = __builtin_amdgcn_wmma_f32_16x16x4_f32(false, a, false, bf,
                                                (short)0, c, false, false);
    }
    if (col < F_) {
      const int fp = col;
      const float mn = sMean[wv][fp];
      #pragma unroll
      for (int i = 0; i < 8; ++i) {
        const int o = tile * 16 + ((ln < 16) ? i : (8 + i));
        if (o < T_) {
          const float rec = sigmoid_(c[i] + mn);           // reconst[fp][o]
          if (sHeld[wv][fp][o] > 0.0f) {
            const float d = rec - sX[wv][fp][o];
            ls += d * d;
            lc += 1.0f;
          }
        }
      }
    }
  }

  sRed[tid] = ls; __syncthreads();
  for (int s = 64; s > 0; s >>= 1) { if (tid < s) sRed[tid] += sRed[tid + s]; __syncthreads(); }
  if (tid == 0) atomicAdd(&accum[0], sRed[0]);
  __syncthreads();
  sRed[tid] = lc; __syncthreads();
  for (int s = 64; s > 0; s >>= 1) { if (tid < s) sRed[tid] += sRed[tid + s]; __syncthreads(); }
  if (tid == 0) atomicAdd(&accum[1], sRed[0]);
}

extern "C" void kernel_launch(void* const* d_in, const int* in_sizes, int n_in,
                              void* d_out, int out_size, void* d_ws, size_t ws_size,
                              hipStream_t stream) {
  (void)in_sizes; (void)n_in; (void)out_size; (void)ws_size;
  const float* x     = (const float*)d_in[0];
  // d_in[1] water_mask: unused by the reference (all ones)
  const float* obs   = (const float*)d_in[2];
  const float* rnd   = (const float*)d_in[3];
  const float* ker_i = (const float*)d_in[4];
  const float* cci_i = (const float*)d_in[5];
  const float* ker_r = (const float*)d_in[6];
  const float* cci_r = (const float*)d_in[7];
  float* out   = (float*)d_out;
  float* accum = (float*)d_ws;

  zero_accum_k<<<1, 1, 0, stream>>>(accum);
  interp_k<<<N_ / WPB, 128, 0, stream>>>(x, obs, ker_i, cci_i, out);
  recon_k<<<N_ / WPB, 128, 0, stream>>>(x, obs, rnd, ker_r, cci_r, accum);
  finalize_k<<<1, 1, 0, stream>>>(accum, out + Y_ELEMS);
}